// KplanesEncoder_14680198218027
// MI455X (gfx1250) — compile-verified
//
#include <hip/hip_runtime.h>
#include <cstdint>
#include <cstddef>

// ---------------------------------------------------------------------------
// K-planes encoder for MI455X (gfx1250).
//  Pass 1: transpose each plane set (3,C,H,W) -> (3,H,W,C) into d_ws so every
//          bilinear tap (32 ch) is one contiguous 128B line (L2-resident).
//  Pass 2: 8 lanes per point, float4 per lane; 4 levels x 6 planes x 4 taps
//          of coalesced b128 gathers; product composite; NT stores of output.
// ---------------------------------------------------------------------------

static constexpr int NRES = 4;
static constexpr int NCH  = 32;
static constexpr int kS[NRES] = {64, 128, 256, 512};
static constexpr int kT[NRES] = {25, 50, 100, 150};

typedef float f4 __attribute__((ext_vector_type(4)));

// ---- probe gfx1250 async-to-LDS builtins (guarded; safe fallback) ----------
#if defined(__has_builtin)
#  if __has_builtin(__builtin_amdgcn_global_load_async_to_lds_b32)
#    define KP_ASYNC_LDS 1
#  endif
#  if __has_builtin(__builtin_amdgcn_s_wait_asynccnt)
#    define KP_WAIT_ASYNC_BUILTIN 1
#  endif
#endif

#if defined(KP_ASYNC_LDS)
typedef __attribute__((address_space(1))) int kp_gint;   // global addrspace
typedef __attribute__((address_space(3))) int kp_lint;   // LDS addrspace
#endif

struct KPParams {
  const float* sp[NRES];   // spatial plane base per level
  const float* tp[NRES];   // temporal plane base per level
  const float* in_tensor;
  const float* time;
  const float* bbox;
  float* out;
  int npts;
};

__device__ __forceinline__ float kp_clamp01(float v) {
  return __builtin_fminf(__builtin_fmaxf(v, 0.0f), 1.0f);
}

// Bilinear sample of 4 consecutive channels [c0, c0+4) from one plane.
// TR=true : plane layout (H,W,C)  -> four 16B vector loads (coalesced 128B/tap
//           across the 8 lanes covering one point).
// TR=false: plane layout (C,H,W)  -> strided scalar fallback.
template <bool TR>
__device__ __forceinline__ f4 kp_sample(const float* __restrict__ plane,
                                        int H, int W, float u, float v, int c0) {
  const float fx = u * (float)(W - 1);
  const float fy = v * (float)(H - 1);
  const float x0f = floorf(fx);
  const float y0f = floorf(fy);
  const float wx = fx - x0f;
  const float wy = fy - y0f;
  int x0 = (int)x0f; x0 = x0 < 0 ? 0 : x0; x0 = x0 > W - 1 ? W - 1 : x0;
  int y0 = (int)y0f; y0 = y0 < 0 ? 0 : y0; y0 = y0 > H - 1 ? H - 1 : y0;
  const int x1 = (x0 + 1 < W) ? x0 + 1 : W - 1;
  const int y1 = (y0 + 1 < H) ? y0 + 1 : H - 1;
  const float w00 = (1.0f - wx) * (1.0f - wy);
  const float w01 = wx * (1.0f - wy);
  const float w10 = (1.0f - wx) * wy;
  const float w11 = wx * wy;

  if constexpr (TR) {
    const f4 a = *(const f4*)(plane + ((size_t)y0 * W + x0) * NCH + c0);
    const f4 b = *(const f4*)(plane + ((size_t)y0 * W + x1) * NCH + c0);
    const f4 c = *(const f4*)(plane + ((size_t)y1 * W + x0) * NCH + c0);
    const f4 d = *(const f4*)(plane + ((size_t)y1 * W + x1) * NCH + c0);
    return a * w00 + b * w01 + c * w10 + d * w11;
  } else {
    const size_t HW  = (size_t)H * W;
    const size_t i00 = (size_t)y0 * W + x0;
    const size_t i01 = (size_t)y0 * W + x1;
    const size_t i10 = (size_t)y1 * W + x0;
    const size_t i11 = (size_t)y1 * W + x1;
    f4 r;
#pragma unroll
    for (int k = 0; k < 4; ++k) {
      const float* pc = plane + (size_t)(c0 + k) * HW;
      r[k] = pc[i00] * w00 + pc[i01] * w01 + pc[i10] * w10 + pc[i11] * w11;
    }
    return r;
  }
}

// ---- Pass 1: (3,C,H,W) -> (3,H,W,C), fully coalesced both directions -------
__global__ __launch_bounds__(256) void kp_transpose(const float* __restrict__ in,
                                                    float* __restrict__ out,
                                                    int H, int W) {
  __shared__ float tile[NCH][NCH + 1];   // +1 pad: conflict-free transpose
  const int p   = blockIdx.z;
  const int y   = blockIdx.y;
  const int x0  = blockIdx.x * 32;
  const int tid = threadIdx.x;
  const int lx  = tid & 31;
  const int r8  = tid >> 5;
  const size_t HW = (size_t)H * W;

  // Reads of the original CHW planes are last-use: non-temporal so they don't
  // occupy L2 that we want for the transposed copy.
  const float* ip = in + (size_t)p * NCH * HW + (size_t)y * W + x0;
#pragma unroll
  for (int r = 0; r < 4; ++r) {
    const int c = r * 8 + r8;
    tile[c][lx] = __builtin_nontemporal_load(ip + (size_t)c * HW + lx);
  }
  __syncthreads();
  float* op = out + ((size_t)p * HW + (size_t)y * W + x0) * NCH;
#pragma unroll
  for (int r = 0; r < 4; ++r) {
    const int xo = r * 8 + r8;
    op[(size_t)xo * NCH + lx] = tile[lx][xo];
  }
}

// ---- Pass 2: main gather/interp kernel -------------------------------------
template <bool TR>
__global__ __launch_bounds__(256) void kp_main(KPParams P) {
  const int tid = threadIdx.x;
  const int gid = blockIdx.x * 256 + tid;
  const int pid = gid >> 3;          // 8 lanes per point
  const int c0  = (gid & 7) * 4;     // this lane's 4 channels
  const int blockP0 = (blockIdx.x * 256) >> 3;   // 32 points per block

  // Stage the block's 32 points (96 coord floats + 32 time floats) into LDS
  // via gfx1250 async global->LDS loads when available (ASYNCcnt path).
  __shared__ float sc[128];
  {
    const int np3 = P.npts * 3;
#if defined(KP_ASYNC_LDS)
    if (tid < 96) {
      const int gi = blockP0 * 3 + tid;
      if (gi < np3)
        __builtin_amdgcn_global_load_async_to_lds_b32(
            (kp_gint*)(P.in_tensor + gi), (kp_lint*)&sc[tid], 0, 0);
    } else if (tid < 128) {
      const int gi = blockP0 + (tid - 96);
      if (gi < P.npts)
        __builtin_amdgcn_global_load_async_to_lds_b32(
            (kp_gint*)(P.time + gi), (kp_lint*)&sc[tid], 0, 0);
    }
    if (tid < 128) {
#  if defined(KP_WAIT_ASYNC_BUILTIN)
      __builtin_amdgcn_s_wait_asynccnt(0);
#  else
      asm volatile("s_wait_asynccnt 0" ::: "memory");
#  endif
    }
#else
    if (tid < 96) {
      const int gi = blockP0 * 3 + tid;
      sc[tid] = (gi < np3) ? P.in_tensor[gi] : 0.0f;
    } else if (tid < 128) {
      const int gi = blockP0 + (tid - 96);
      sc[tid] = (gi < P.npts) ? P.time[gi] : 0.0f;
    }
#endif
  }
  __syncthreads();
  if (pid >= P.npts) return;

  const int lp = tid >> 3;   // local point id: LDS broadcast across 8 lanes
  const float t = sc[96 + lp];
  float q[3];
#pragma unroll
  for (int k = 0; k < 3; ++k) {
    const float lo = P.bbox[k];
    const float hi = P.bbox[3 + k];
    q[k] = kp_clamp01((sc[lp * 3 + k] - lo) / (hi - lo));
  }

  float* outp = P.out + (size_t)pid * (NRES * NCH);

#pragma unroll
  for (int i = 0; i < NRES; ++i) {
    const int Si = kS[i], Ti = kT[i];
    const size_t sStr = (size_t)NCH * Si * Si;  // plane stride (both layouts)
    const size_t tStr = (size_t)NCH * Ti * Si;

    f4 acc;
    // spatial planes: (x,y), (y,z), (x,z); u -> W axis, v -> H axis
    acc  = kp_sample<TR>(P.sp[i] + 0 * sStr, Si, Si, q[0], q[1], c0);
    acc *= kp_sample<TR>(P.sp[i] + 1 * sStr, Si, Si, q[1], q[2], c0);
    acc *= kp_sample<TR>(P.sp[i] + 2 * sStr, Si, Si, q[0], q[2], c0);
    // temporal planes: (x,t), (y,t), (z,t); W = spatial axis, H = time axis
    acc *= kp_sample<TR>(P.tp[i] + 0 * tStr, Ti, Si, q[0], t, c0);
    acc *= kp_sample<TR>(P.tp[i] + 1 * tStr, Ti, Si, q[1], t, c0);
    acc *= kp_sample<TR>(P.tp[i] + 2 * tStr, Ti, Si, q[2], t, c0);

    // Output is a 256MB one-shot stream: NT store keeps the 168MB of
    // transposed planes resident in the 192MB L2.
    __builtin_nontemporal_store(acc, (f4*)(outp + i * NCH + c0));
  }
}

// ---------------------------------------------------------------------------
extern "C" void kernel_launch(void* const* d_in, const int* in_sizes, int n_in,
                              void* d_out, int out_size, void* d_ws, size_t ws_size,
                              hipStream_t stream) {
  (void)n_in; (void)out_size;

  KPParams P;
  P.in_tensor = (const float*)d_in[0];
  P.time      = (const float*)d_in[1];
  P.bbox      = (const float*)d_in[2];
  P.out       = (float*)d_out;
  P.npts      = in_sizes[0] / 3;

  // dict order: spatial_emb{i}, temporal_emb{i} interleaved
  const float* sp_raw[NRES];
  const float* tp_raw[NRES];
  for (int i = 0; i < NRES; ++i) {
    sp_raw[i] = (const float*)d_in[3 + 2 * i];
    tp_raw[i] = (const float*)d_in[4 + 2 * i];
  }

  // Workspace layout for the transposed (H,W,C) copies.
  size_t spOff[NRES], tpOff[NRES], total = 0;
  for (int i = 0; i < NRES; ++i) {
    spOff[i] = total; total += (size_t)3 * NCH * kS[i] * kS[i];
    tpOff[i] = total; total += (size_t)3 * NCH * kT[i] * kS[i];
  }
  const bool useTr = (ws_size >= total * sizeof(float));

  const int nthreads = P.npts * 8;
  const int blocks = (nthreads + 255) / 256;

  if (useTr) {
    float* ws = (float*)d_ws;
    for (int i = 0; i < NRES; ++i) {
      dim3 gs(kS[i] / 32, kS[i], 3);
      kp_transpose<<<gs, dim3(256), 0, stream>>>(sp_raw[i], ws + spOff[i], kS[i], kS[i]);
      dim3 gt(kS[i] / 32, kT[i], 3);
      kp_transpose<<<gt, dim3(256), 0, stream>>>(tp_raw[i], ws + tpOff[i], kT[i], kS[i]);
      P.sp[i] = ws + spOff[i];
      P.tp[i] = ws + tpOff[i];
    }
    kp_main<true><<<dim3(blocks), dim3(256), 0, stream>>>(P);
  } else {
    for (int i = 0; i < NRES; ++i) { P.sp[i] = sp_raw[i]; P.tp[i] = tp_raw[i]; }
    kp_main<false><<<dim3(blocks), dim3(256), 0, stream>>>(P);
  }
}